// random_drop_17798344474844
// MI455X (gfx1250) — compile-verified
//
#include <hip/hip_runtime.h>

// Problem shape (from reference): x[B=16, S=8192, D=256] f32, idx[B,2048] int,
// out[b, j, :] = x[b, idx[b,j], :]  -> out[16, 2048, 256] f32.
//
// Pure gather: 1 KB contiguous per gathered row. Memory-bound:
// ~64 MiB HBM traffic -> ~2.8 us floor at 23.3 TB/s. Strategy:
//   - one float4 (16 B) per thread: global_load_b128 / global_store_b128
//   - row index is wave32-uniform (64 float4 per row = 2 waves per row),
//     so scalarize it via readfirstlane -> s_load_b32 on the SMEM path
//   - perfectly coalesced stores; 512 B contiguous per-wave loads
//   - speculative global_prefetch_b8 on upcoming index cachelines

constexpr int kB   = 16;
constexpr int kS   = 8192;
constexpr int kD   = 256;
constexpr int kRef = 2048;

constexpr unsigned kVecPerRow = kD / 4;                         // 64 float4 per row
constexpr unsigned kRows      = (unsigned)kB * kRef;            // 32768 rows
constexpr unsigned kTotalVec  = kRows * kVecPerRow;             // 2,097,152 float4
constexpr unsigned kBlock     = 256;                            // 8 wave32s

__global__ __launch_bounds__(kBlock) void gather_rows_b128_kernel(
    const float4* __restrict__ x,      // [B * S * 64] float4
    const int*    __restrict__ idx,    // [B * 2048]
    float4*       __restrict__ out) {  // [B * 2048 * 64] float4
  const unsigned t = blockIdx.x * kBlock + threadIdx.x;  // flat float4 index
  const unsigned row = t >> 6;          // b*2048 + j ; uniform across a wave32
  const unsigned c   = t & 63u;         // float4 column within the row
  const unsigned b   = row >> 11;       // row / 2048

  // Row id is wave-uniform: force scalar (SMEM) load of the gather index.
  const unsigned rowU = (unsigned)__builtin_amdgcn_readfirstlane((int)row);
  const int s = idx[rowU];

  // Speculative prefetch of index data a few cachelines ahead (OOB-safe by mask).
  __builtin_prefetch(idx + ((rowU + 256u) & (kRows - 1u)), 0, 0);

  // Source: x[b, s, c*4 .. c*4+3] as one float4.
  const size_t src = ((size_t)b * kS + (size_t)(unsigned)s) * kVecPerRow + c;

  out[t] = x[src];   // global_load_b128 -> global_store_b128
}

extern "C" void kernel_launch(void* const* d_in, const int* in_sizes, int n_in,
                              void* d_out, int out_size, void* d_ws, size_t ws_size,
                              hipStream_t stream) {
  (void)in_sizes; (void)n_in; (void)out_size; (void)d_ws; (void)ws_size;

  const float4* x   = (const float4*)d_in[0];
  const int*    idx = (const int*)d_in[1];
  float4*       out = (float4*)d_out;

  const unsigned blocks = kTotalVec / kBlock;   // 8192 blocks x 256 threads
  gather_rows_b128_kernel<<<dim3(blocks), dim3(kBlock), 0, stream>>>(x, idx, out);
}